// HilbertLayer_4844723109893
// MI455X (gfx1250) — compile-verified
//
#include <hip/hip_runtime.h>
#include <stdint.h>

// Pointer payload type the gfx1250 async global<->LDS builtins expect
// (from hipcc diagnostic: 'int __vector_size__(4*sizeof(int)) *').
typedef int   v4i __attribute__((vector_size(16)));
// Native float vector for the nontemporal fallback (HIP_vector_type is
// rejected by __builtin_nontemporal_*).
typedef float v4f __attribute__((vector_size(16)));

#if __has_builtin(__builtin_amdgcn_global_load_async_to_lds_b128) && \
    __has_builtin(__builtin_amdgcn_global_store_async_from_lds_b128)
#define HAVE_ASYNC_LDS 1
#else
#define HAVE_ASYNC_LDS 0
#endif

__device__ __forceinline__ void wait_asynccnt0() {
#if __has_builtin(__builtin_amdgcn_s_wait_asynccnt)
    __builtin_amdgcn_s_wait_asynccnt(0);
#else
    asm volatile("s_wait_asynccnt 0x0" ::: "memory");
#endif
}

// Shapes fixed by the reference: B=32, H=W=64, C=256.
// HW = 4096 rows per batch, 64 float4 per row (1 KiB rows).
// Total float4 elements = 32*4096*64 = 8,388,608 -> fits in uint32 math.
//
// Each block: 256 threads = 4 rows staged through 4 KiB of LDS via the
// CDNA5 async copy path (ASYNCcnt). A wave32 covers half a row, so the
// Hilbert index d is wave-uniform -> scalar idx load via readfirstlane.
__global__ __launch_bounds__(256, 8)
void hilbert_gather_async(const float4* __restrict__ x4,
                          const int*    __restrict__ idx,
                          float4*       __restrict__ out4,
                          unsigned int n4)
{
    __shared__ float4 stage[256];

    const unsigned int tid = threadIdx.x;
    const unsigned int e   = blockIdx.x * 256u + tid;   // global float4 index
    if (e >= n4) return;

    const unsigned int lane = e & 63u;     // float4 within the 1 KiB row
    const unsigned int row  = e >> 6;      // = b*4096 + d   (wave-uniform)
    const unsigned int d    = row & 4095u; // Hilbert position (wave-uniform)

    // Force d into an SGPR so idx[d] becomes a scalar (K$) load.
    const int du = __builtin_amdgcn_readfirstlane((int)d);
    const unsigned int src_row = (unsigned int)idx[du];

    // b*4096 == row - d ; source element index within x.
    const unsigned int src_e = (row - d + src_row) * 64u + lane;

    const float4* src = x4   + src_e;
    float4*       dst = out4 + e;

#if HAVE_ASYNC_LDS
    // global -> LDS, tracked by ASYNCcnt (gfx1250 async data-mover path).
    __builtin_amdgcn_global_load_async_to_lds_b128(
        (v4i*)src, (v4i*)&stage[tid], /*imm offset*/0, /*cpol*/0);

    // Wave-local RAW: each wave stores exactly the LDS bytes it loaded,
    // so draining this wave's ASYNCcnt is sufficient (no s_barrier needed).
    wait_asynccnt0();

    // LDS -> global; consecutive lanes write contiguous 512 B per wave.
    __builtin_amdgcn_global_store_async_from_lds_b128(
        (v4i*)dst, (v4i*)&stage[tid], /*imm offset*/0, /*cpol*/0);
    // S_ENDPGM performs an implicit wait-idle, draining the async stores.
#else
    // Fallback: direct streaming copy (still fully coalesced, NT hints to
    // reduce L2 pollution of the 256 MiB streamed working set).
    v4f v = __builtin_nontemporal_load((const v4f*)src);
    __builtin_nontemporal_store(v, (v4f*)dst);
    (void)stage;
#endif
}

extern "C" void kernel_launch(void* const* d_in, const int* in_sizes, int n_in,
                              void* d_out, int out_size, void* d_ws, size_t ws_size,
                              hipStream_t stream) {
    (void)n_in; (void)d_ws; (void)ws_size; (void)in_sizes;

    const float* x   = (const float*)d_in[0];   // [B,H,W,C] f32
    const int*   idx = (const int*)d_in[1];     // [H*W] i32
    float*       out = (float*)d_out;           // [B,1,H*W,C] f32

    const unsigned int n4     = (unsigned int)(out_size / 4); // float4 count
    const unsigned int blocks = (n4 + 255u) / 256u;

    hilbert_gather_async<<<blocks, 256, 0, stream>>>(
        (const float4*)x, idx, (float4*)out, n4);
}